// GNN_64510408786469
// MI455X (gfx1250) — compile-verified
//
#include <hip/hip_runtime.h>

// ---------------------------------------------------------------------------
// 3-layer GraphSAGE on MI455X (gfx1250, wave32).
// Phase 1 per layer: segment mean via float atomics (memory-bound, ~1.1 GB
// total traffic -> ~50us at 23.3 TB/s; this is the roofline floor).
// Phase 2 per layer: fused dual-GEMM  out = agg@Wl + x_tgt@Wr + b  using
// V_WMMA_F32_16X16X4_F32 (fp32 matches reference; GEMMs are only ~19 GFLOP
// total, far from being the bottleneck, weights fit in L2).
// ---------------------------------------------------------------------------

typedef float v2f __attribute__((ext_vector_type(2)));
typedef float v8f __attribute__((ext_vector_type(8)));

__global__ void zero_f32(float* __restrict__ p, int n) {
    int i = blockIdx.x * blockDim.x + threadIdx.x;
    if (i < n) p[i] = 0.0f;
}

__global__ void edge_count(const int* __restrict__ dst, float* __restrict__ cnt, int E) {
    int e = blockIdx.x * blockDim.x + threadIdx.x;
    if (e < E) atomicAdd(&cnt[dst[e]], 1.0f);
}

// One thread per (edge, 4-float chunk): gather float4 from source row,
// scatter-add into the destination segment sum.
__global__ void edge_scatter(const float* __restrict__ xs, const int* __restrict__ src,
                             const int* __restrict__ dst, float* __restrict__ sum,
                             int E, int D) {
    int chunks = D >> 2;
    long long tid = (long long)blockIdx.x * blockDim.x + threadIdx.x;
    if (tid >= (long long)E * chunks) return;
    int e = (int)(tid / chunks);
    int c = (int)(tid % chunks);
    const float4 v = *(const float4*)(xs + (size_t)src[e] * D + (size_t)c * 4);
    float* s = sum + (size_t)dst[e] * D + (size_t)c * 4;
    atomicAdd(s + 0, v.x);
    atomicAdd(s + 1, v.y);
    atomicAdd(s + 2, v.z);
    atomicAdd(s + 3, v.w);
}

__global__ void seg_normalize(float* __restrict__ sum, const float* __restrict__ cnt,
                              int n, int D) {
    int chunks = D >> 2;
    long long tid = (long long)blockIdx.x * blockDim.x + threadIdx.x;
    if (tid >= (long long)n * chunks) return;
    int i = (int)(tid / chunks);
    int c = (int)(tid % chunks);
    float inv = 1.0f / fmaxf(cnt[i], 1.0f);
    float4* p = (float4*)(sum + (size_t)i * D + (size_t)c * 4);
    float4 v = *p;
    v.x *= inv; v.y *= inv; v.z *= inv; v.w *= inv;
    *p = v;
}

// Fused dual GEMM: out[M,N] = A0[M,K]@W0[K,N] + A1[M,K]@W1[K,N] + bias, opt ReLU.
// One wave per 16x16 output tile; K-loop in steps of 4 with f32 WMMA.
// A-matrix lane layout (ISA 7.12.2, 32-bit A 16x4):
//   lanes 0-15: row=lane,   K = k0+{0,1};  lanes 16-31: row=lane-16, K = k0+{2,3}
// B-matrix (4x16, rows striped across lanes):
//   lanes 0-15: col=lane,   K = k0+{0,1};  lanes 16-31: col=lane-16, K = k0+{2,3}
// C/D: VGPR v -> M = v + (lane>=16 ? 8 : 0), N = lane&15.
__global__ void sage_gemm_wmma(const float* __restrict__ A0,  // agg   M x K
                               const float* __restrict__ A1,  // x_tgt M x K
                               const float* __restrict__ W0,  // Wl    K x N
                               const float* __restrict__ W1,  // Wr    K x N
                               const float* __restrict__ bias,
                               float* __restrict__ out,       // M x N
                               int M, int K, int N, int relu) {
    int wave = blockIdx.x * (blockDim.x >> 5) + (threadIdx.x >> 5);
    int tilesN = N >> 4;
    int totalTiles = (M >> 4) * tilesN;
    if (wave >= totalTiles) return;   // uniform per-wave: EXEC stays all-1s

    int tm = wave / tilesN;
    int tn = wave - tm * tilesN;
    int lane = threadIdx.x & 31;
    int half = lane >> 4;             // 0: K-pair {0,1}; 1: K-pair {2,3}
    int l15  = lane & 15;
    int arow = (tm << 4) + l15;
    int bcol = (tn << 4) + l15;

    v8f acc = {0.f, 0.f, 0.f, 0.f, 0.f, 0.f, 0.f, 0.f};

    #pragma unroll
    for (int pass = 0; pass < 2; ++pass) {
        const float* A = pass ? A1 : A0;
        const float* W = pass ? W1 : W0;
        const float* ap = A + (size_t)arow * K + (half << 1);
        #pragma unroll 4
        for (int k0 = 0; k0 < K; k0 += 4) {
            v2f a = *(const v2f*)(ap + k0);       // 8B aligned: k0%4==0, half*2 even
            int bk = k0 + (half << 1);
            v2f b;
            b.x = W[(size_t)bk * N + bcol];
            b.y = W[(size_t)(bk + 1) * N + bcol];
            acc = __builtin_amdgcn_wmma_f32_16x16x4_f32(
                /*neg_a=*/false, a, /*neg_b=*/false, b,
                /*c_mod=*/(short)0, acc, /*reuse_a=*/false, /*reuse_b=*/false);
        }
    }

    float bv = bias[bcol];
    float* op = out + (size_t)((tm << 4) + (half << 3)) * N + bcol;
    #pragma unroll
    for (int v = 0; v < 8; ++v) {
        float r = acc[v] + bv;
        if (relu) r = fmaxf(r, 0.0f);
        op[(size_t)v * N] = r;
    }
}

extern "C" void kernel_launch(void* const* d_in, const int* in_sizes, int n_in,
                              void* d_out, int out_size, void* d_ws, size_t ws_size,
                              hipStream_t stream) {
    (void)in_sizes; (void)n_in; (void)out_size; (void)ws_size;

    const float* x = (const float*)d_in[0];
    const int* srcp[3] = {(const int*)d_in[1], (const int*)d_in[3], (const int*)d_in[5]};
    const int* dstp[3] = {(const int*)d_in[2], (const int*)d_in[4], (const int*)d_in[6]};
    const float* Wl[3] = {(const float*)d_in[7],  (const float*)d_in[10], (const float*)d_in[13]};
    const float* Wr[3] = {(const float*)d_in[8],  (const float*)d_in[11], (const float*)d_in[14]};
    const float* bb[3] = {(const float*)d_in[9],  (const float*)d_in[12], (const float*)d_in[15]};

    const int Ecnt[3] = {1500000, 300000, 60000};
    const int ntgt[3] = {100000, 20000, 4000};
    const int Kd[3]   = {128, 256, 256};
    const int Nd[3]   = {256, 256, 128};
    const int relu[3] = {1, 0, 1};   // reference applies relu when i != 1

    // Workspace layout (floats)
    float* ws     = (float*)d_ws;
    float* aggbuf = ws;                      // max n_tgt*K = 100000*128 = 12.8M
    float* cntbuf = aggbuf + 12800000;       // max n_tgt = 100000
    float* h0     = cntbuf + 100000;         // 100000*256 = 25.6M
    float* h1     = h0 + 25600000;           // 20000*256  = 5.12M

    const float* layer_in[3] = {x, h0, h1};
    float* layer_out[3]      = {h0, h1, (float*)d_out};

    const int TPB = 256;
    for (int L = 0; L < 3; ++L) {
        int E = Ecnt[L], n = ntgt[L], K = Kd[L], N = Nd[L];

        int zn = n * K;
        zero_f32<<<(zn + TPB - 1) / TPB, TPB, 0, stream>>>(aggbuf, zn);
        zero_f32<<<(n + TPB - 1) / TPB, TPB, 0, stream>>>(cntbuf, n);

        edge_count<<<(E + TPB - 1) / TPB, TPB, 0, stream>>>(dstp[L], cntbuf, E);

        long long sthreads = (long long)E * (K >> 2);
        edge_scatter<<<(unsigned)((sthreads + TPB - 1) / TPB), TPB, 0, stream>>>(
            layer_in[L], srcp[L], dstp[L], aggbuf, E, K);

        long long nthreads = (long long)n * (K >> 2);
        seg_normalize<<<(unsigned)((nthreads + TPB - 1) / TPB), TPB, 0, stream>>>(
            aggbuf, cntbuf, n, K);

        int tiles = (n >> 4) * (N >> 4);
        int wavesPerBlock = TPB >> 5;  // 8 waves (wave32)
        sage_gemm_wmma<<<(tiles + wavesPerBlock - 1) / wavesPerBlock, TPB, 0, stream>>>(
            aggbuf, layer_in[L], Wl[L], Wr[L], bb[L], layer_out[L], n, K, N, relu[L]);
    }
}